// Attn_8134668058995
// MI455X (gfx1250) — compile-verified
//
#include <hip/hip_runtime.h>
#include <hip/hip_bf16.h>

typedef float v2f __attribute__((ext_vector_type(2)));
typedef float v4f __attribute__((ext_vector_type(4)));
typedef float v8f __attribute__((ext_vector_type(8)));

#define HIDDEN  512
#define MAXLEN  4096
#define BATCH   32

// ---------------------------------------------------------------------------
// Kernel 1: Q[32x512] = hidden[32x512] @ W[512x512]   (q[b,h] = sum_o h[b,o]W[o,h])
// One wave per 16x16 output tile, K stepped by 4 via v_wmma_f32_16x16x4_f32.
// 2 (M tiles) x 32 (N tiles) = 64 tiles = 8 blocks * 8 waves, EXEC always full.
// ---------------------------------------------------------------------------
__global__ void __launch_bounds__(256) qgemm_wmma(const float* __restrict__ hidden,
                                                  const float* __restrict__ W,
                                                  float* __restrict__ q) {
    const int lane = threadIdx.x & 31;
    const int tile = blockIdx.x * 8 + (threadIdx.x >> 5);  // 0..63
    const int mTile = tile & 1;                            // batch tile (32/16)
    const int nTile = tile >> 1;                           // h tile (512/16)
    const int m0 = mTile * 16;
    const int n0 = nTile * 16;
    const int half = lane >> 4;                            // 0: lanes 0-15, 1: lanes 16-31
    const int lid  = lane & 15;

    v8f acc = {0.f, 0.f, 0.f, 0.f, 0.f, 0.f, 0.f, 0.f};

    // A[m][k] layout: lane half selects K-pair; row m = lid
    const float* aRow = hidden + (size_t)(m0 + lid) * HIDDEN + half * 2;
    // B[k][n] = W[k*512 + n]; lane half selects K-pair; col n = lid
    const float* bCol = W + n0 + lid;

    for (int k0 = 0; k0 < HIDDEN; k0 += 4) {
        v2f a;
        a.x = aRow[k0 + 0];
        a.y = aRow[k0 + 1];
        const float* bp = bCol + (size_t)(k0 + half * 2) * HIDDEN;
        v2f bm;
        bm.x = bp[0];
        bm.y = bp[HIDDEN];
        // (neg_a, A, neg_b, B, c_mod, C, reuse_a, reuse_b)
        acc = __builtin_amdgcn_wmma_f32_16x16x4_f32(false, a, false, bm,
                                                    (short)0, acc, false, false);
    }

    // D layout: VGPR r -> row m = r + 8*half, col n = lid
    float* qout = q + n0 + lid;
#pragma unroll
    for (int r = 0; r < 8; ++r) {
        const int m = m0 + r + half * 8;
        qout[(size_t)m * HIDDEN] = acc[r];
    }
}

// ---------------------------------------------------------------------------
// Kernel 2: energies[b,l] = Q[b] . enc[l,b]   (HBM-bound: 256 MB streamed once)
// One wave per (l,b) dot product; b128 loads, non-temporal on encoder stream.
// Block of 8 waves covers 8 consecutive b of one l -> 16 KB contiguous.
// ---------------------------------------------------------------------------
__global__ void __launch_bounds__(256) energies_kernel(const float* __restrict__ enc,
                                                       const float* __restrict__ q,
                                                       float* __restrict__ energies) {
    const int lane = threadIdx.x & 31;
    const int w = blockIdx.x * 8 + (threadIdx.x >> 5);   // global wave id
    const int b = w & 31;
    const int l = w >> 5;

    const v4f* e4 = (const v4f*)(enc + (size_t)(l * BATCH + b) * HIDDEN);
    const v4f* q4 = (const v4f*)(q + (size_t)b * HIDDEN);

    float acc = 0.f;
#pragma unroll
    for (int i = 0; i < 4; ++i) {
        v4f ev = __builtin_nontemporal_load(&e4[i * 32 + lane]);  // stream, don't cache
        v4f qv = q4[i * 32 + lane];                               // hot, stays in L2
        acc = fmaf(ev.x, qv.x, acc);
        acc = fmaf(ev.y, qv.y, acc);
        acc = fmaf(ev.z, qv.z, acc);
        acc = fmaf(ev.w, qv.w, acc);
    }
    // wave32 reduction
#pragma unroll
    for (int m = 16; m > 0; m >>= 1) acc += __shfl_xor(acc, m, 32);
    if (lane == 0) energies[(size_t)b * MAXLEN + l] = acc;
}

// ---------------------------------------------------------------------------
// Kernel 3: row softmax over l (bias term cancels under the softmax shift).
// One block per batch row; 16 elements per thread held in registers.
// ---------------------------------------------------------------------------
__global__ void __launch_bounds__(256) softmax_kernel(const float* __restrict__ energies,
                                                      float* __restrict__ out) {
    __shared__ float red[256];
    const int b = blockIdx.x;
    const int t = threadIdx.x;
    const float* row = energies + (size_t)b * MAXLEN;

    float vals[16];
    float mx = -3.402823466e38f;
#pragma unroll
    for (int i = 0; i < 16; ++i) {
        vals[i] = row[t + i * 256];
        mx = fmaxf(mx, vals[i]);
    }
    red[t] = mx;
    __syncthreads();
    for (int s = 128; s > 0; s >>= 1) {
        if (t < s) red[t] = fmaxf(red[t], red[t + s]);
        __syncthreads();
    }
    mx = red[0];
    __syncthreads();

    float sum = 0.f;
#pragma unroll
    for (int i = 0; i < 16; ++i) {
        vals[i] = __expf(vals[i] - mx);
        sum += vals[i];
    }
    red[t] = sum;
    __syncthreads();
    for (int s = 128; s > 0; s >>= 1) {
        if (t < s) red[t] += red[t + s];
        __syncthreads();
    }
    const float inv = 1.0f / red[0];
#pragma unroll
    for (int i = 0; i < 16; ++i) {
        out[(size_t)b * MAXLEN + t + i * 256] = vals[i] * inv;
    }
}

// ---------------------------------------------------------------------------
extern "C" void kernel_launch(void* const* d_in, const int* in_sizes, int n_in,
                              void* d_out, int out_size, void* d_ws, size_t ws_size,
                              hipStream_t stream) {
    (void)in_sizes; (void)n_in; (void)out_size; (void)ws_size;
    const float* hidden = (const float*)d_in[0];   // [1, 32, 512]
    const float* enc    = (const float*)d_in[1];   // [4096, 32, 512]
    const float* W      = (const float*)d_in[2];   // [512, 512]
    // d_in[3] = bias: mathematically cancels in softmax (constant shift per row).

    float* q        = (float*)d_ws;                                    // 64 KB
    float* energies = (float*)((char*)d_ws + BATCH * HIDDEN * sizeof(float)); // 512 KB

    qgemm_wmma<<<8, 256, 0, stream>>>(hidden, W, q);
    energies_kernel<<<(MAXLEN * BATCH) / 8, 256, 0, stream>>>(enc, q, energies);
    softmax_kernel<<<BATCH, 256, 0, stream>>>(energies, (float*)d_out);
}